// GNN_15333033247248
// MI455X (gfx1250) — compile-verified
//
#include <hip/hip_runtime.h>
#include <hip/hip_bf16.h>

typedef __attribute__((ext_vector_type(2))) float v2f;
typedef __attribute__((ext_vector_type(8))) float v8f;

#define N_NODES 100000
#define N_EDGES 1000000
#define HID 64
#define N_LAYERS 4
#define LPITCH 68   // LDS row pitch in floats (bank-conflict-free b64 frag loads)

// ---------------------------------------------------------------------------
// k_prep: v1 = W^T a1, v2 = W^T a2, c1 = b.a1, c2 = b.a2   (one tiny block)
// vbuf layout: [0..63]=v1, [64..127]=v2, [128]=c1, [129]=c2
// ---------------------------------------------------------------------------
__global__ void k_prep(const float* __restrict__ W, const float* __restrict__ b,
                       const float* __restrict__ attW, float* __restrict__ vbuf) {
    int k = threadIdx.x;  // 64 threads
    float s1 = 0.f, s2 = 0.f;
    for (int h = 0; h < HID; ++h) {
        float w = W[h * HID + k];
        s1 = fmaf(w, attW[h], s1);
        s2 = fmaf(w, attW[HID + h], s2);
    }
    vbuf[k] = s1;
    vbuf[HID + k] = s2;
    if (k == 0) {
        float c1 = 0.f, c2 = 0.f;
        for (int h = 0; h < HID; ++h) {
            c1 = fmaf(b[h], attW[h], c1);
            c2 = fmaf(b[h], attW[HID + h], c2);
        }
        vbuf[128] = c1;
        vbuf[129] = c2;
    }
}

// ---------------------------------------------------------------------------
// k_psd: per node n: psrc[n] = x[n].v1 + c1 ; pdst[n] = x[n].v2 + c2 ; deg[n]=0
// 16 lanes per node, float4 loads, shuffle reduction within 16-lane group.
// ---------------------------------------------------------------------------
__global__ __launch_bounds__(256) void k_psd(const float* __restrict__ xin,
                                             const float* __restrict__ vbuf,
                                             float* __restrict__ psrc,
                                             float* __restrict__ pdst,
                                             float* __restrict__ deg) {
    int tid = threadIdx.x;
    int g = tid >> 4;          // group 0..15 (one node each)
    int j = tid & 15;          // lane within group
    int node = blockIdx.x * 16 + g;
    if (node >= N_NODES) return;

    float4 xv = *(const float4*)(xin + (size_t)node * HID + 4 * j);
    float4 w1 = *(const float4*)(vbuf + 4 * j);
    float4 w2 = *(const float4*)(vbuf + HID + 4 * j);
    float d1 = xv.x * w1.x + xv.y * w1.y + xv.z * w1.z + xv.w * w1.w;
    float d2 = xv.x * w2.x + xv.y * w2.y + xv.z * w2.z + xv.w * w2.w;
    d1 += __shfl_xor(d1, 1); d1 += __shfl_xor(d1, 2);
    d1 += __shfl_xor(d1, 4); d1 += __shfl_xor(d1, 8);
    d2 += __shfl_xor(d2, 1); d2 += __shfl_xor(d2, 2);
    d2 += __shfl_xor(d2, 4); d2 += __shfl_xor(d2, 8);
    if (j == 0) {
        psrc[node] = d1 + vbuf[128];
        pdst[node] = d2 + vbuf[129];
        deg[node]  = 0.0f;
    }
}

// ---------------------------------------------------------------------------
// k_edge: a_e = exp(leaky_relu(psrc[src]+pdst[dst])); atomic add into deg[dst]
// ---------------------------------------------------------------------------
__global__ __launch_bounds__(256) void k_edge(const int* __restrict__ e,
                                              const float* __restrict__ psrc,
                                              const float* __restrict__ pdst,
                                              float* __restrict__ deg) {
    int i = blockIdx.x * 256 + threadIdx.x;
    if (i >= N_EDGES) return;
    int2 ed = ((const int2*)e)[i];
    float l = psrc[ed.x] + pdst[ed.y];
    l = (l > 0.0f) ? l : 0.2f * l;     // leaky_relu, slope 0.2
    float a = expf(l);
    unsafeAtomicAdd(&deg[ed.y], a);    // global_atomic_add_f32
}

// ---------------------------------------------------------------------------
// k_node: 128 nodes / block (256 thr = 8 waves).
//  phase1: xmid = rmsnorm(relu(x/deg') + x, nw, nb)  -> LDS
//  phase2: per wave, y-tile = xmid(16x64) @ hW^T via V_WMMA_F32_16X16X4_F32,
//          w = xmid + relu(y + hb)  -> LDS (in place)
//  phase3: out = rmsnorm(w, hnw, hnb) -> global
// ---------------------------------------------------------------------------
__global__ __launch_bounds__(256) void k_node(const float* __restrict__ xin,
                                              const float* __restrict__ deg,
                                              const float* __restrict__ nw,
                                              const float* __restrict__ nb,
                                              const float* __restrict__ hW,
                                              const float* __restrict__ hb,
                                              const float* __restrict__ hnw,
                                              const float* __restrict__ hnb,
                                              float* __restrict__ xout) {
    __shared__ __align__(16) float sx[128 * LPITCH];  // xmid then w
    __shared__ __align__(16) float sw[HID * LPITCH];  // hid_W (row-major, padded)

    int tid = threadIdx.x;
    int g = tid >> 4;       // 16-lane group id (0..15)
    int j = tid & 15;       // lane within group
    int base = blockIdx.x * 128;

    // stage hid_W into LDS (4096 floats, float4 per thread x4)
    for (int idx = tid; idx < 1024; idx += 256) {
        int h = idx >> 4;
        int c = (idx & 15) << 2;
        float4 v = *(const float4*)(hW + h * HID + c);
        *(float4*)(&sw[h * LPITCH + c]) = v;
    }

    // ---- phase 1 ----
    float4 nwv = *(const float4*)(nw + 4 * j);
    float4 nbv = *(const float4*)(nb + 4 * j);
#pragma unroll
    for (int i = 0; i < 8; ++i) {
        int nl = g + (i << 4);
        int node = base + nl;
        float4 z;
        if (node < N_NODES) {
            float4 xv = *(const float4*)(xin + (size_t)node * HID + 4 * j);
            float d = deg[node];
            d = (d == 0.0f) ? 1.0f : d;
            float id = 1.0f / d;
            z.x = fmaxf(xv.x * id, 0.0f) + xv.x;
            z.y = fmaxf(xv.y * id, 0.0f) + xv.y;
            z.z = fmaxf(xv.z * id, 0.0f) + xv.z;
            z.w = fmaxf(xv.w * id, 0.0f) + xv.w;
        } else {
            z.x = z.y = z.z = z.w = 0.0f;
        }
        float ss = z.x * z.x + z.y * z.y + z.z * z.z + z.w * z.w;
        ss += __shfl_xor(ss, 1); ss += __shfl_xor(ss, 2);
        ss += __shfl_xor(ss, 4); ss += __shfl_xor(ss, 8);
        float inv = rsqrtf(ss * (1.0f / 64.0f) + 1e-5f);
        float4 xm;
        xm.x = nwv.x * (z.x * inv) + nbv.x;
        xm.y = nwv.y * (z.y * inv) + nbv.y;
        xm.z = nwv.z * (z.z * inv) + nbv.z;
        xm.w = nwv.w * (z.w * inv) + nbv.w;
        *(float4*)(&sx[nl * LPITCH + 4 * j]) = xm;
    }
    __syncthreads();

    // ---- phase 2: WMMA f32 16x16x4, K=64 over 16 steps, 4 N-tiles ----
    int lane = tid & 31;
    int wave = tid >> 5;              // 0..7, each owns 16 nodes
    int tileM = wave << 4;
    int mrow = lane & 15;             // A row (node) / B col (h within tile)
    int koff = (lane >> 4) << 1;      // lanes 16-31 take K+2,K+3
    const float* arow = &sx[(tileM + mrow) * LPITCH + koff];
    const float* b0r = &sw[(0  + mrow) * LPITCH + koff];
    const float* b1r = &sw[(16 + mrow) * LPITCH + koff];
    const float* b2r = &sw[(32 + mrow) * LPITCH + koff];
    const float* b3r = &sw[(48 + mrow) * LPITCH + koff];

    v8f acc[4];
#pragma unroll
    for (int t = 0; t < 4; ++t) acc[t] = (v8f){0,0,0,0,0,0,0,0};

#pragma unroll
    for (int k = 0; k < HID; k += 4) {
        v2f af  = *(const v2f*)(arow + k);
        v2f bf0 = *(const v2f*)(b0r + k);
        v2f bf1 = *(const v2f*)(b1r + k);
        v2f bf2 = *(const v2f*)(b2r + k);
        v2f bf3 = *(const v2f*)(b3r + k);
        acc[0] = __builtin_amdgcn_wmma_f32_16x16x4_f32(false, af, false, bf0, (short)0, acc[0], false, false);
        acc[1] = __builtin_amdgcn_wmma_f32_16x16x4_f32(false, af, false, bf1, (short)0, acc[1], false, false);
        acc[2] = __builtin_amdgcn_wmma_f32_16x16x4_f32(false, af, false, bf2, (short)0, acc[2], false, false);
        acc[3] = __builtin_amdgcn_wmma_f32_16x16x4_f32(false, af, false, bf3, (short)0, acc[3], false, false);
    }

    // w = xmid + relu(y + hb): C/D layout -> VGPR r holds M=r (lanes 0-15) / M=8+r
    int mhi = (lane >> 4) << 3;
    int hl = lane & 15;
#pragma unroll
    for (int t = 0; t < 4; ++t) {
        int h = (t << 4) + hl;
        float hbv = hb[h];
#pragma unroll
        for (int r = 0; r < 8; ++r) {
            float y = fmaxf(acc[t][r] + hbv, 0.0f);
            sx[(tileM + mhi + r) * LPITCH + h] += y;
        }
    }
    __syncthreads();

    // ---- phase 3 ----
    float4 hnwv = *(const float4*)(hnw + 4 * j);
    float4 hnbv = *(const float4*)(hnb + 4 * j);
#pragma unroll
    for (int i = 0; i < 8; ++i) {
        int nl = g + (i << 4);
        int node = base + nl;
        float4 wv = *(const float4*)(&sx[nl * LPITCH + 4 * j]);
        float ss = wv.x * wv.x + wv.y * wv.y + wv.z * wv.z + wv.w * wv.w;
        ss += __shfl_xor(ss, 1); ss += __shfl_xor(ss, 2);
        ss += __shfl_xor(ss, 4); ss += __shfl_xor(ss, 8);
        float inv = rsqrtf(ss * (1.0f / 64.0f) + 1e-5f);
        float4 o;
        o.x = hnwv.x * (wv.x * inv) + hnbv.x;
        o.y = hnwv.y * (wv.y * inv) + hnbv.y;
        o.z = hnwv.z * (wv.z * inv) + hnbv.z;
        o.w = hnwv.w * (wv.w * inv) + hnbv.w;
        if (node < N_NODES)
            *(float4*)(xout + (size_t)node * HID + 4 * j) = o;
    }
}

// ---------------------------------------------------------------------------
extern "C" void kernel_launch(void* const* d_in, const int* in_sizes, int n_in,
                              void* d_out, int out_size, void* d_ws, size_t ws_size,
                              hipStream_t stream) {
    const float* x      = (const float*)d_in[0];
    const int*   e      = (const int*)d_in[1];
    const float* convW  = (const float*)d_in[2];
    const float* convB  = (const float*)d_in[3];
    const float* attW   = (const float*)d_in[4];
    const float* convNW = (const float*)d_in[5];
    const float* convNB = (const float*)d_in[6];
    const float* hidW   = (const float*)d_in[7];
    const float* hidB   = (const float*)d_in[8];
    const float* hidNW  = (const float*)d_in[9];
    const float* hidNB  = (const float*)d_in[10];
    float* out = (float*)d_out;

    // workspace layout
    char* ws = (char*)d_ws;
    float* F    = (float*)ws;                                     // 100000*64 f32
    float* psrc = (float*)(ws + (size_t)N_NODES * HID * 4);       // +25.6MB
    float* pdst = psrc + N_NODES;
    float* deg  = pdst + N_NODES;
    float* vbuf = deg + N_NODES;                                  // 130 floats

    const float* xin = x;
    for (int l = 0; l < N_LAYERS; ++l) {
        float* xout = (l & 1) ? out : F;
        if (l == N_LAYERS - 1) xout = out;   // final layer -> d_out (l=3 odd anyway)

        k_prep<<<1, 64, 0, stream>>>(convW + (size_t)l * HID * HID,
                                     convB + (size_t)l * HID,
                                     attW + (size_t)l * 2 * HID, vbuf);
        k_psd<<<(N_NODES + 15) / 16, 256, 0, stream>>>(xin, vbuf, psrc, pdst, deg);
        k_edge<<<(N_EDGES + 255) / 256, 256, 0, stream>>>(e, psrc, pdst, deg);
        k_node<<<(N_NODES + 127) / 128, 256, 0, stream>>>(
            xin, deg,
            convNW + (size_t)l * HID, convNB + (size_t)l * HID,
            hidW + (size_t)l * HID * HID, hidB + (size_t)l * HID,
            hidNW + (size_t)l * HID, hidNB + (size_t)l * HID, xout);
        xin = xout;
    }
}